// GraphConvolution_82179904241991
// MI455X (gfx1250) — compile-verified
//
#include <hip/hip_runtime.h>

typedef __attribute__((ext_vector_type(2))) float v2f;
typedef __attribute__((ext_vector_type(8))) float v8f;

#define N_NODES 100000
#define N_EDGES 1600000
#define IN_F    256
#define OUT_F   64

// ---------------------------------------------------------------------------
// Kernel 1: support = X @ W  (fp32, V_WMMA_F32_16X16X4_F32)
// 256 threads = 8 waves. Wave w computes rows [blk*128 + w*16, +16), N=0..63.
// W (256x64 f32 = 64KB) staged in LDS with a 16-bank rotation per K-pair
// (rows k and k+2 land 16 banks apart -> no half-wave bank conflicts).
// A-frag layout (ISA 7.12.2, 32-bit A 16x4): lane lo=M, v0=K(2*hi), v1=K(2*hi+1).
// B-frag layout (4x16, row-per-half-VGPR): v0=B[k0+2*hi][n], v1=B[k0+2*hi+1][n].
// C/D layout: lane(lo,hi), VGPR r -> D[r + 8*hi][lo].
// ---------------------------------------------------------------------------
__global__ __launch_bounds__(256) void gemm_xw_wmma(const float* __restrict__ X,
                                                    const float* __restrict__ W,
                                                    float* __restrict__ support) {
  __shared__ float lw[IN_F * OUT_F];  // 64 KB, swizzled

  // Cooperative load of W with bank-rotation swizzle:
  // element (k, n) stored at k*64 + (((n>>4) + (k>>1)) & 3)*16 + (n&15)
  for (int i = threadIdx.x; i < IN_F * OUT_F; i += 256) {
    int k = i >> 6;
    int n = i & 63;
    int sl = k * 64 + ((((n >> 4) + (k >> 1)) & 3) << 4) + (n & 15);
    lw[sl] = W[i];
  }
  __syncthreads();

  const int wave = threadIdx.x >> 5;
  const int lane = threadIdx.x & 31;
  const int hi   = lane >> 4;      // half-wave select
  const int lo   = lane & 15;
  const int row_base = blockIdx.x * 128 + wave * 16;
  if (row_base >= N_NODES) return;  // wave-uniform: EXEC stays all-ones

  const float* xrow = X + (size_t)(row_base + lo) * IN_F;

  v8f acc0 = {}, acc1 = {}, acc2 = {}, acc3 = {};
  for (int k0 = 0; k0 < IN_F; k0 += 4) {
    const int ka = k0 + 2 * hi;               // K row pair (ka, ka+1)
    v2f a = *(const v2f*)(xrow + ka);         // A fragment: contiguous float2/lane
    const float* wr = lw + ka * 64;
    const int kr = ka >> 1;
    const int s0 = (((kr + 0) & 3) << 4) + lo;
    const int s1 = (((kr + 1) & 3) << 4) + lo;
    const int s2 = (((kr + 2) & 3) << 4) + lo;
    const int s3 = (((kr + 3) & 3) << 4) + lo;
    v2f b0 = { wr[s0], wr[s0 + 64] };   // cols  0..15
    v2f b1 = { wr[s1], wr[s1 + 64] };   // cols 16..31
    v2f b2 = { wr[s2], wr[s2 + 64] };   // cols 32..47
    v2f b3 = { wr[s3], wr[s3 + 64] };   // cols 48..63
    acc0 = __builtin_amdgcn_wmma_f32_16x16x4_f32(false, a, false, b0, (short)0, acc0, false, false);
    acc1 = __builtin_amdgcn_wmma_f32_16x16x4_f32(false, a, false, b1, (short)0, acc1, false, false);
    acc2 = __builtin_amdgcn_wmma_f32_16x16x4_f32(false, a, false, b2, (short)0, acc2, false, false);
    acc3 = __builtin_amdgcn_wmma_f32_16x16x4_f32(false, a, false, b3, (short)0, acc3, false, false);
  }

  // Store D: VGPR r of accN -> support[row_base + 8*hi + r][ntile*16 + lo]
  float* orow = support + (size_t)(row_base + 8 * hi) * OUT_F + lo;
#pragma unroll
  for (int r = 0; r < 8; ++r) {
    orow[(size_t)r * OUT_F +  0] = acc0[r];
    orow[(size_t)r * OUT_F + 16] = acc1[r];
    orow[(size_t)r * OUT_F + 32] = acc2[r];
    orow[(size_t)r * OUT_F + 48] = acc3[r];
  }
}

// ---------------------------------------------------------------------------
// Kernel 2: out[n][f] = bias[f]   (float4 broadcast, grid-stride)
// ---------------------------------------------------------------------------
__global__ __launch_bounds__(256) void init_bias(const float* __restrict__ bias,
                                                 float* __restrict__ out) {
  const float4* b4 = (const float4*)bias;   // 16 x float4
  float4* o4 = (float4*)out;
  const int total = N_NODES * OUT_F / 4;    // 1,600,000
  for (int i = blockIdx.x * blockDim.x + threadIdx.x; i < total;
       i += gridDim.x * blockDim.x) {
    o4[i] = b4[i & 15];
  }
}

// ---------------------------------------------------------------------------
// Kernel 3: out[row[e]] += edge_vals[e] * support[col[e]]   (rows sorted)
// One wave owns 128 consecutive edges; lane owns output columns (2*lane, 2*lane+1).
// Edges processed in groups of 4: int4/float4 index+value loads, then 4
// independent support gathers issued back-to-back (MLP ~4) before the
// (wave-uniform) row-change reduction logic. Flush with global_atomic_add_f32
// only on row change; avg 16 edges/row -> ~9 flushes per 128 edges.
// N_EDGES (1.6M) is divisible by 128, so chunks and int4 loads are exact/aligned.
// ---------------------------------------------------------------------------
#define EDGES_PER_WAVE 128

__global__ __launch_bounds__(256) void spmm_scatter(const float* __restrict__ support,
                                                    const float* __restrict__ edge_vals,
                                                    const int* __restrict__ row_idx,
                                                    const int* __restrict__ col_idx,
                                                    float* __restrict__ out) {
  const int lane = threadIdx.x & 31;
  const int wave = blockIdx.x * 8 + (threadIdx.x >> 5);
  const long e0 = (long)wave * EDGES_PER_WAVE;
  if (e0 >= N_EDGES) return;

  const int c2 = lane * 2;
  float ax = 0.f, ay = 0.f;
  int cur = -1;

  for (long e = e0; e < e0 + EDGES_PER_WAVE; e += 4) {
    const int4   rr = *(const int4*)(row_idx + e);
    const int4   cc = *(const int4*)(col_idx + e);
    const float4 vv = *(const float4*)(edge_vals + e);
    // Issue all 4 gathers before consuming any (independent loads in flight)
    const float2 s0 = *(const float2*)(support + (size_t)cc.x * OUT_F + c2);
    const float2 s1 = *(const float2*)(support + (size_t)cc.y * OUT_F + c2);
    const float2 s2 = *(const float2*)(support + (size_t)cc.z * OUT_F + c2);
    const float2 s3 = *(const float2*)(support + (size_t)cc.w * OUT_F + c2);

    if (rr.x != cur) {
      if (cur >= 0) {
        atomicAdd(out + (size_t)cur * OUT_F + c2,     ax);
        atomicAdd(out + (size_t)cur * OUT_F + c2 + 1, ay);
      }
      cur = rr.x; ax = 0.f; ay = 0.f;
    }
    ax = fmaf(vv.x, s0.x, ax); ay = fmaf(vv.x, s0.y, ay);

    if (rr.y != cur) {
      atomicAdd(out + (size_t)cur * OUT_F + c2,     ax);
      atomicAdd(out + (size_t)cur * OUT_F + c2 + 1, ay);
      cur = rr.y; ax = 0.f; ay = 0.f;
    }
    ax = fmaf(vv.y, s1.x, ax); ay = fmaf(vv.y, s1.y, ay);

    if (rr.z != cur) {
      atomicAdd(out + (size_t)cur * OUT_F + c2,     ax);
      atomicAdd(out + (size_t)cur * OUT_F + c2 + 1, ay);
      cur = rr.z; ax = 0.f; ay = 0.f;
    }
    ax = fmaf(vv.z, s2.x, ax); ay = fmaf(vv.z, s2.y, ay);

    if (rr.w != cur) {
      atomicAdd(out + (size_t)cur * OUT_F + c2,     ax);
      atomicAdd(out + (size_t)cur * OUT_F + c2 + 1, ay);
      cur = rr.w; ax = 0.f; ay = 0.f;
    }
    ax = fmaf(vv.w, s3.x, ax); ay = fmaf(vv.w, s3.y, ay);
  }

  if (cur >= 0) {
    atomicAdd(out + (size_t)cur * OUT_F + c2,     ax);
    atomicAdd(out + (size_t)cur * OUT_F + c2 + 1, ay);
  }
}

// ---------------------------------------------------------------------------
extern "C" void kernel_launch(void* const* d_in, const int* in_sizes, int n_in,
                              void* d_out, int out_size, void* d_ws, size_t ws_size,
                              hipStream_t stream) {
  const float* X         = (const float*)d_in[0];
  const float* edge_vals = (const float*)d_in[1];
  const float* W         = (const float*)d_in[2];
  const float* bias      = (const float*)d_in[3];
  const int*   row_idx   = (const int*)d_in[4];
  const int*   col_idx   = (const int*)d_in[5];
  float* out     = (float*)d_out;
  float* support = (float*)d_ws;   // needs 100000*64*4 = 25.6 MB

  (void)in_sizes; (void)n_in; (void)out_size; (void)ws_size;

  // 1) support = X @ W   (WMMA fp32)
  gemm_xw_wmma<<<(N_NODES + 127) / 128, 256, 0, stream>>>(X, W, support);
  // 2) out = bias (broadcast)
  init_bias<<<2048, 256, 0, stream>>>(bias, out);
  // 3) out += A @ support (edge scatter with register segment-reduction)
  int waves  = (N_EDGES + EDGES_PER_WAVE - 1) / EDGES_PER_WAVE;  // 12500
  int blocks = (waves + 7) / 8;
  spmm_scatter<<<blocks, 256, 0, stream>>>(support, edge_vals, row_idx, col_idx, out);
}